// PointNet2_59820304499055
// MI455X (gfx1250) — compile-verified
//
#include <hip/hip_runtime.h>

// ---------------------------------------------------------------------------
// PointNet++ forward for MI455X (gfx1250, wave32).
//   - sa1/sa2 feature MLPs are dead code in the reference -> only FPS runs.
//   - FPS1: xyz (48MB) is L2-resident (192MB L2); dist kept in workspace.
//   - All activations stored point-major [P][C] so GEMM A and B operands are
//     K-contiguous: inner loop = five global_load_b64 + four
//     V_WMMA_F32_16X16X4_F32, branch-free (K padded to multiple of 4).
//   - One wave computes a 16(M) x 64(N) tile: 4 independent accumulator
//     chains + register double-buffer (prefetch next K-step before issuing
//     WMMAs) so load latency overlaps the WMMA chains instead of stalling
//     at s_wait_loadcnt 0.
// ---------------------------------------------------------------------------

typedef __attribute__((ext_vector_type(2))) float v2f;
typedef __attribute__((ext_vector_type(4))) float v4f;
typedef __attribute__((ext_vector_type(8))) float v8f;

#define BB    64
#define NPTS  65536
#define SAMP1 512
#define SAMP2 128
#define PPTS  (BB * SAMP2)   // 8192 points through sa3 MLP

// ---------------- FPS over the large point cloud (dist in global ws) -------
__global__ void fps_large_kernel(const float* __restrict__ xyz,
                                 float* __restrict__ dist_ws,
                                 float* __restrict__ out_xyz,
                                 int N, int S)
{
  const int b  = blockIdx.x;
  const int t  = threadIdx.x;
  const int nt = blockDim.x;
  const float* __restrict__ px = xyz + (size_t)b * N * 3;
  float* __restrict__ dist = dist_ws + (size_t)b * N;

  __shared__ float s_val[1024];
  __shared__ int   s_idx[1024];
  __shared__ float s_cent[3];

  for (int i = t; i < N; i += nt) dist[i] = 1e10f;
  if (t == 0) { s_cent[0] = px[0]; s_cent[1] = px[1]; s_cent[2] = px[2]; }
  __syncthreads();

  for (int s = 0; s < S; ++s) {
    const float cx = s_cent[0], cy = s_cent[1], cz = s_cent[2];
    if (t == 0) {  // record sampled point (scan emits previous 'far')
      float* o = out_xyz + ((size_t)b * S + s) * 3;
      o[0] = cx; o[1] = cy; o[2] = cz;
    }
    float bestd = -1.0f; int besti = 0;
    for (int i = t; i < N; i += nt) {
      float dx = px[3*i+0] - cx;
      float dy = px[3*i+1] - cy;
      float dz = px[3*i+2] - cz;
      float d  = dx*dx + dy*dy + dz*dz;
      float dm = fminf(dist[i], d);
      dist[i]  = dm;
      if (dm > bestd) { bestd = dm; besti = i; }
    }
    s_val[t] = bestd; s_idx[t] = besti;
    __syncthreads();
    for (int off = nt >> 1; off > 0; off >>= 1) {
      if (t < off && s_val[t + off] > s_val[t]) {
        s_val[t] = s_val[t + off]; s_idx[t] = s_idx[t + off];
      }
      __syncthreads();
    }
    if (t == 0) {
      int fi = s_idx[0];
      s_cent[0] = px[3*fi+0]; s_cent[1] = px[3*fi+1]; s_cent[2] = px[3*fi+2];
    }
    __syncthreads();
  }
}

// ---------------- FPS over 512 points, fully LDS-resident ------------------
__global__ void fps_small_kernel(const float* __restrict__ xyz,
                                 float* __restrict__ out_xyz,
                                 int N, int S)
{
  const int b = blockIdx.x;
  const int t = threadIdx.x;          // one thread per point, blockDim == N
  __shared__ float sx[512], sy[512], sz[512];
  __shared__ float s_val[512];
  __shared__ int   s_idx[512];
  __shared__ float s_cent[3];

  const float* __restrict__ px = xyz + (size_t)b * N * 3;
  sx[t] = px[3*t+0]; sy[t] = px[3*t+1]; sz[t] = px[3*t+2];
  float dist = 1e10f;
  if (t == 0) { s_cent[0] = sx[0]; s_cent[1] = sy[0]; s_cent[2] = sz[0]; }
  __syncthreads();

  for (int s = 0; s < S; ++s) {
    const float cx = s_cent[0], cy = s_cent[1], cz = s_cent[2];
    if (t == 0) {
      float* o = out_xyz + ((size_t)b * S + s) * 3;
      o[0] = cx; o[1] = cy; o[2] = cz;
    }
    float dx = sx[t]-cx, dy = sy[t]-cy, dz = sz[t]-cz;
    dist = fminf(dist, dx*dx + dy*dy + dz*dz);
    s_val[t] = dist; s_idx[t] = t;
    __syncthreads();
    for (int off = 256; off > 0; off >>= 1) {
      if (t < off && s_val[t + off] > s_val[t]) {
        s_val[t] = s_val[t + off]; s_idx[t] = s_idx[t + off];
      }
      __syncthreads();
    }
    if (t == 0) { int fi = s_idx[0]; s_cent[0]=sx[fi]; s_cent[1]=sy[fi]; s_cent[2]=sz[fi]; }
    __syncthreads();
  }
}

// ---------------- Layer-1 K-padding prep kernels ---------------------------
// X0[n][0..3] = {xyz2[n].x, .y, .z, 0}   (point-major, K padded 3 -> 4)
__global__ void pad_x0_kernel(const float* __restrict__ xyz2,
                              float* __restrict__ X0, int P)
{
  const int n = blockIdx.x * blockDim.x + threadIdx.x;
  if (n >= P) return;
  v4f v;
  v.x = xyz2[3*n+0]; v.y = xyz2[3*n+1]; v.z = xyz2[3*n+2]; v.w = 0.0f;
  *(v4f*)(X0 + 4*(size_t)n) = v;
}
// W1p[m][0..3] = {W1[m][0..2], 0}
__global__ void pad_w1_kernel(const float* __restrict__ W1,
                              float* __restrict__ W1p, int M)
{
  const int m = blockIdx.x * blockDim.x + threadIdx.x;
  if (m >= M) return;
  v4f v;
  v.x = W1[3*m+0]; v.y = W1[3*m+1]; v.z = W1[3*m+2]; v.w = 0.0f;
  *(v4f*)(W1p + 4*(size_t)m) = v;
}

// ---------------- f32 GEMM via V_WMMA_F32_16X16X4_F32 ----------------------
// Y(m,n) = sum_k A[m*lda + k] * B[n*ldb + k] + bias        (K % 4 == 0)
// One wave computes a 16(M) x 64(N) tile: 4 accumulator chains, software
// pipelined (next K-step's fragments loaded before current WMMAs issue).
// A frag: lanes0-15 hold (k0,k0+1) of row tm+l; lanes16-31 hold (k0+2,k0+3).
// C/D:    VGPR r -> row tm + r + half*8, col tn + l.
// out_nmajor=1: Y(m,n) at Y[n*ldo + m], bias per-m, vectorized b128 stores.
// out_nmajor=0: Y(m,n) at Y[m*ldo + n], bias per-n.
__global__ void wmma_gemm_f32(const float* __restrict__ A, int lda,
                              const float* __restrict__ Bp, int ldb,
                              const float* __restrict__ bias,
                              float* __restrict__ Y,
                              int M, int N, int K, int ldo, int out_nmajor)
{
  const int wave = (blockIdx.x * blockDim.x + threadIdx.x) >> 5;
  const int lane = threadIdx.x & 31;
  const int ntn  = N >> 6;                 // 64-wide N tiles
  const int tm   = (wave / ntn) << 4;
  const int tn   = (wave % ntn) << 6;
  if (tm >= M) return;

  const int half = lane >> 4;              // selects K pair within a K4 step
  const int l    = lane & 15;

  const float* __restrict__ Arow = A  + (size_t)(tm + l) * lda;
  const float* __restrict__ B0   = Bp + (size_t)(tn + l) * ldb;

  v8f acc[4];
#pragma unroll
  for (int j = 0; j < 4; ++j) acc[j] = (v8f){};

  // prologue: load first K-step fragments
  v2f a_cur = *(const v2f*)(Arow + half * 2);
  v2f b_cur[4];
#pragma unroll
  for (int j = 0; j < 4; ++j)
    b_cur[j] = *(const v2f*)(B0 + (size_t)(j * 16) * ldb + half * 2);

  // steady state: prefetch k0+4 while the 4 WMMA chains consume k0
  for (int k0 = 0; k0 + 4 < K; k0 += 4) {
    const int kn = k0 + 4 + half * 2;
    const v2f a_nxt = *(const v2f*)(Arow + kn);
    v2f b_nxt[4];
#pragma unroll
    for (int j = 0; j < 4; ++j)
      b_nxt[j] = *(const v2f*)(B0 + (size_t)(j * 16) * ldb + kn);
#pragma unroll
    for (int j = 0; j < 4; ++j)
      acc[j] = __builtin_amdgcn_wmma_f32_16x16x4_f32(false, a_cur, false, b_cur[j],
                                                     (short)0, acc[j], false, false);
    a_cur = a_nxt;
#pragma unroll
    for (int j = 0; j < 4; ++j) b_cur[j] = b_nxt[j];
  }
  // epilogue K-step
#pragma unroll
  for (int j = 0; j < 4; ++j)
    acc[j] = __builtin_amdgcn_wmma_f32_16x16x4_f32(false, a_cur, false, b_cur[j],
                                                   (short)0, acc[j], false, false);

  if (out_nmajor) {                        // activations: point-major [N][M]
#pragma unroll
    for (int j = 0; j < 4; ++j) {
      float* p = Y + (size_t)(tn + j * 16 + l) * ldo + tm + half * 8;
      v4f lo, hi;
#pragma unroll
      for (int r = 0; r < 4; ++r) {
        lo[r] = acc[j][r]     + bias[tm + half * 8 + r];
        hi[r] = acc[j][r + 4] + bias[tm + half * 8 + 4 + r];
      }
      *(v4f*)(p)     = lo;
      *(v4f*)(p + 4) = hi;
    }
  } else {                                 // FC: row-major [M][N]
#pragma unroll
    for (int j = 0; j < 4; ++j) {
      const int col = tn + j * 16 + l;
      const float bv = bias[col];
#pragma unroll
      for (int r = 0; r < 8; ++r)
        Y[(size_t)(tm + r + half * 8) * ldo + col] = acc[j][r] + bv;
    }
  }
}

// ---------------- Per-channel batch statistics (channel-fastest layout) ----
// val(c, i) = Y[i*C + c], reduce over i in [0, count)
__global__ void bn_stats_kernel(const float* __restrict__ Y,
                                float* __restrict__ mean,
                                float* __restrict__ rstd,
                                int C, int count)
{
  const int c = blockIdx.x;
  float s = 0.0f, ss = 0.0f;
  for (int i = threadIdx.x; i < count; i += blockDim.x) {
    float v = Y[(size_t)i * C + c];
    s += v; ss += v * v;
  }
  __shared__ float rs[256], rss[256];
  rs[threadIdx.x] = s; rss[threadIdx.x] = ss;
  __syncthreads();
  for (int off = blockDim.x >> 1; off > 0; off >>= 1) {
    if (threadIdx.x < off) {
      rs[threadIdx.x]  += rs[threadIdx.x + off];
      rss[threadIdx.x] += rss[threadIdx.x + off];
    }
    __syncthreads();
  }
  if (threadIdx.x == 0) {
    float m   = rs[0] / (float)count;
    float var = rss[0] / (float)count - m * m;
    mean[c] = m;
    rstd[c] = rsqrtf(var + 1e-5f);
  }
}

// ---------------- Normalize + affine + ReLU (flat, channel-fastest) --------
__global__ void bn_apply_kernel(float* __restrict__ Y,
                                const float* __restrict__ mean,
                                const float* __restrict__ rstd,
                                const float* __restrict__ g,
                                const float* __restrict__ be,
                                int C, long long total)
{
  const long long idx = (long long)blockIdx.x * blockDim.x + threadIdx.x;
  if (idx >= total) return;
  const int c = (int)(idx & (long long)(C - 1));   // C is a power of two
  float v = (Y[idx] - mean[c]) * rstd[c] * g[c] + be[c];
  Y[idx] = v > 0.0f ? v : 0.0f;
}

// ---------------- Per-batch max-pool over points ---------------------------
// H: [B*Mpts][C] point-major; f3: [B][C] row-major (feeds FC gemm directly)
__global__ void maxpool_kernel(const float* __restrict__ H,
                               float* __restrict__ f3,
                               int C, int B, int Mpts)
{
  const int idx = blockIdx.x * blockDim.x + threadIdx.x;
  if (idx >= B * C) return;
  const int b = idx / C, c = idx & (C - 1);
  const float* __restrict__ p = H + (size_t)b * Mpts * C + c;
  float m = p[0];
  for (int i = 1; i < Mpts; ++i) m = fmaxf(m, p[(size_t)i * C]);
  f3[idx] = m;
}

// ---------------- Final 256 -> 1 projection --------------------------------
__global__ void final_fc3_kernel(const float* __restrict__ X,
                                 const float* __restrict__ W,
                                 const float* __restrict__ b0,
                                 float* __restrict__ out, int C)
{
  const int b = blockIdx.x;
  float s = 0.0f;
  for (int i = threadIdx.x; i < C; i += blockDim.x) s += X[(size_t)b * C + i] * W[i];
  __shared__ float rs[256];
  rs[threadIdx.x] = s;
  __syncthreads();
  for (int off = blockDim.x >> 1; off > 0; off >>= 1) {
    if (threadIdx.x < off) rs[threadIdx.x] += rs[threadIdx.x + off];
    __syncthreads();
  }
  if (threadIdx.x == 0) out[b] = rs[0] + b0[0];
}

// ---------------------------------------------------------------------------
extern "C" void kernel_launch(void* const* d_in, const int* in_sizes, int n_in,
                              void* d_out, int out_size, void* d_ws, size_t ws_size,
                              hipStream_t stream) {
  (void)in_sizes; (void)n_in; (void)out_size; (void)ws_size;

  // Input leaf order: xyz, sa1(12), sa2(12), sa3(12), fc1(4), fc2(4), fc3(2)
  const float* xyz = (const float*)d_in[0];
  const float* W1  = (const float*)d_in[25]; const float* b1  = (const float*)d_in[26];
  const float* g1  = (const float*)d_in[27]; const float* be1 = (const float*)d_in[28];
  const float* W2  = (const float*)d_in[29]; const float* b2  = (const float*)d_in[30];
  const float* g2  = (const float*)d_in[31]; const float* be2 = (const float*)d_in[32];
  const float* W3  = (const float*)d_in[33]; const float* b3  = (const float*)d_in[34];
  const float* g3  = (const float*)d_in[35]; const float* be3 = (const float*)d_in[36];
  const float* Wf1 = (const float*)d_in[37]; const float* bf1 = (const float*)d_in[38];
  const float* gf1 = (const float*)d_in[39]; const float* bef1= (const float*)d_in[40];
  const float* Wf2 = (const float*)d_in[41]; const float* bf2 = (const float*)d_in[42];
  const float* gf2 = (const float*)d_in[43]; const float* bef2= (const float*)d_in[44];
  const float* Wf3 = (const float*)d_in[45]; const float* bf3 = (const float*)d_in[46];
  float* out = (float*)d_out;

  // Workspace arena (lifetimes overlapped): dist [0,16MB) dies before H3 [0,32MB)
  char* ws = (char*)d_ws;
  float* dist = (float*)(ws);                       // B*N      = 16 MB (FPS1 only)
  float* H3   = (float*)(ws);                       // P x 1024 = 32 MB
  float* H2   = (float*)(ws + ((size_t)32 << 20));  // P x 512  = 16 MB
  float* H1   = (float*)(ws + ((size_t)48 << 20));  // P x 256  =  8 MB
  char*  sm   = ws + ((size_t)56 << 20);
  float* xyz1 = (float*)sm; sm += (size_t)BB * SAMP1 * 3 * 4;
  float* xyz2 = (float*)sm; sm += (size_t)BB * SAMP2 * 3 * 4;
  float* X0   = (float*)sm; sm += (size_t)PPTS * 4 * 4;      // padded layer1 input
  float* W1p  = (float*)sm; sm += (size_t)256 * 4 * 4;       // padded layer1 weight
  float* f3   = (float*)sm; sm += (size_t)BB * 1024 * 4;
  float* X1   = (float*)sm; sm += (size_t)BB * 512 * 4;
  float* X2   = (float*)sm; sm += (size_t)BB * 256 * 4;
  float* mean = (float*)sm; sm += 1024 * 4;
  float* rstd = (float*)sm;

  // ---- sa1 / sa2: FPS only (features are dead code in the reference) ----
  fps_large_kernel<<<BB, 1024, 0, stream>>>(xyz, dist, xyz1, NPTS, SAMP1);
  fps_small_kernel<<<BB, SAMP1, 0, stream>>>(xyz1, xyz2, SAMP1, SAMP2);

  // ---- layer-1 K padding (3 -> 4) ----------------------------------------
  pad_x0_kernel<<<PPTS / 256, 256, 0, stream>>>(xyz2, X0, PPTS);
  pad_w1_kernel<<<1, 256, 0, stream>>>(W1, W1p, 256);

  // ---- sa3 MLP over P = 8192 points (activations point-major [P][C]) -----
  {  // layer1: H1[P][256] = (W1p @ X0^T)^T
    int M = 256, N = PPTS, K = 4;
    int waves = (M / 16) * (N / 64);
    wmma_gemm_f32<<<waves / 8, 256, 0, stream>>>(W1p, 4, X0, 4, b1, H1,
                                                 M, N, K, M, 1);
    bn_stats_kernel<<<256, 256, 0, stream>>>(H1, mean, rstd, 256, PPTS);
    bn_apply_kernel<<<(256 * PPTS) / 256, 256, 0, stream>>>(
        H1, mean, rstd, g1, be1, 256, (long long)256 * PPTS);
  }
  {  // layer2: H2[P][512], A = W2[512][256], B = H1[P][256]
    int M = 512, N = PPTS, K = 256;
    int waves = (M / 16) * (N / 64);
    wmma_gemm_f32<<<waves / 8, 256, 0, stream>>>(W2, 256, H1, 256, b2, H2,
                                                 M, N, K, M, 1);
    bn_stats_kernel<<<512, 256, 0, stream>>>(H2, mean, rstd, 512, PPTS);
    bn_apply_kernel<<<(512 * PPTS) / 256, 256, 0, stream>>>(
        H2, mean, rstd, g2, be2, 512, (long long)512 * PPTS);
  }
  {  // layer3: H3[P][1024], A = W3[1024][512], B = H2[P][512]
    int M = 1024, N = PPTS, K = 512;
    int waves = (M / 16) * (N / 64);
    wmma_gemm_f32<<<waves / 8, 256, 0, stream>>>(W3, 512, H2, 512, b3, H3,
                                                 M, N, K, M, 1);
    bn_stats_kernel<<<1024, 256, 0, stream>>>(H3, mean, rstd, 1024, PPTS);
    bn_apply_kernel<<<(1024 * PPTS) / 256, 256, 0, stream>>>(
        H3, mean, rstd, g3, be3, 1024, (long long)1024 * PPTS);
  }
  // global max pool over the 128 points of each batch -> f3 [B][1024]
  maxpool_kernel<<<(BB * 1024) / 256, 256, 0, stream>>>(H3, f3, 1024, BB, SAMP2);

  // ---- FC head ------------------------------------------------------------
  {  // fc1: X1[64][512] = f3[64][1024] @ Wf1^T   (A = f3, B = Wf1[512][1024])
    int M = BB, N = 512, K = 1024;
    int waves = (M / 16) * (N / 64);
    wmma_gemm_f32<<<waves / 8, 256, 0, stream>>>(f3, 1024, Wf1, 1024, bf1, X1,
                                                 M, N, K, N, 0);
    bn_stats_kernel<<<512, 64, 0, stream>>>(X1, mean, rstd, 512, BB);
    bn_apply_kernel<<<(512 * BB) / 256, 256, 0, stream>>>(
        X1, mean, rstd, gf1, bef1, 512, (long long)512 * BB);
  }
  {  // fc2: X2[64][256] = X1[64][512] @ Wf2^T
    int M = BB, N = 256, K = 512;
    int waves = (M / 16) * (N / 64);
    wmma_gemm_f32<<<waves / 8, 256, 0, stream>>>(X1, 512, Wf2, 512, bf2, X2,
                                                 M, N, K, N, 0);
    bn_stats_kernel<<<256, 64, 0, stream>>>(X2, mean, rstd, 256, BB);
    bn_apply_kernel<<<(256 * BB) / 256, 256, 0, stream>>>(
        X2, mean, rstd, gf2, bef2, 256, (long long)256 * BB);
  }
  // fc3: out[64] = X2 @ Wf3^T + bf3
  final_fc3_kernel<<<BB, 256, 0, stream>>>(X2, Wf3, bf3, out, 256);
}